// SiameseNetwork_17454747091442
// MI455X (gfx1250) — compile-verified
//
#include <hip/hip_runtime.h>
#include <hip/hip_bf16.h>

#if defined(__HIP_DEVICE_COMPILE__) && !__has_builtin(__builtin_amdgcn_wmma_f32_16x16x4_f32)
#error "missing __builtin_amdgcn_wmma_f32_16x16x4_f32 on this toolchain (device pass)"
#endif

typedef __attribute__((ext_vector_type(2))) float v2f;
typedef __attribute__((ext_vector_type(8))) float v8f;

#define FEAT 128

// ---------------------------------------------------------------------------
// Generic GEMM:  Y[nrows x M] = X[nrows x K] @ W[K x M]  (+bias) (+relu)
// One block = 256 threads = 8 waves. Block covers a 16-row strip and 128
// columns (blockIdx.y selects which 128-col group). Each wave owns one 16x16
// output tile and runs the K-loop with V_WMMA_F32_16X16X4_F32.
// ---------------------------------------------------------------------------
__global__ __launch_bounds__(256) void gemm_wmma_f32(
    const float* __restrict__ X, const float* __restrict__ W,
    const float* __restrict__ bias, float* __restrict__ Y,
    int nrows, int K, int M, int do_relu)
{
    __shared__ float xs[16 * 512];   // up to K=512

    const int tid  = threadIdx.x;
    const int wave = tid >> 5;
    const int lane = tid & 31;
    const int half = lane >> 4;      // 0: lanes 0-15, 1: lanes 16-31
    const int l16  = lane & 15;

    const int row0 = blockIdx.x * 16;
    const int col0 = blockIdx.y * 128 + wave * 16;

    // cooperative load of the 16 x K input strip (zero-pad past nrows)
    for (int i = tid; i < 16 * K; i += 256) {
        int r = i / K, c = i - r * K;
        int gr = row0 + r;
        xs[r * K + c] = (gr < nrows) ? X[(size_t)gr * K + c] : 0.0f;
    }
    __syncthreads();

    v8f acc = {0.f, 0.f, 0.f, 0.f, 0.f, 0.f, 0.f, 0.f};

#if defined(__HIP_DEVICE_COMPILE__)
    for (int k0 = 0; k0 < K; k0 += 4) {
        const int ka = k0 + half * 2;
        // A frag (16x4): lane l16 holds row M=l16, K = ka, ka+1
        v2f a;
        a.x = xs[l16 * K + ka];
        a.y = xs[l16 * K + ka + 1];
        // B frag (4x16): lane l16 holds col N=l16, K = ka, ka+1
        const float* wp = W + (size_t)ka * M + col0 + l16;
        v2f b;
        b.x = wp[0];
        b.y = wp[M];
        acc = __builtin_amdgcn_wmma_f32_16x16x4_f32(
            /*neg_a=*/false, a, /*neg_b=*/false, b,
            /*c_mod=*/(short)0, acc, /*reuse_a=*/false, /*reuse_b=*/false);
    }
#endif

    // C/D layout: VGPR r -> (row r, lanes 0-15) and (row r+8, lanes 16-31)
    const int col = col0 + l16;
#pragma unroll
    for (int r = 0; r < 8; ++r) {
        int row = row0 + r + half * 8;
        if (row < nrows) {
            float v = acc[r];
            if (bias)    v += bias[col];
            if (do_relu) v  = fmaxf(v, 0.0f);
            Y[(size_t)row * M + col] = v;
        }
    }
}

// ---------------------------------------------------------------------------
// Degree / dinv
// ---------------------------------------------------------------------------
__global__ void deg_init_kernel(float* __restrict__ deg, int n) {
    int i = blockIdx.x * blockDim.x + threadIdx.x;
    if (i < n) deg[i] = 1.0f;                 // self-loop contribution
}

__global__ void deg_count_kernel(const int* __restrict__ dst,
                                 float* __restrict__ deg, int e) {
    int i = blockIdx.x * blockDim.x + threadIdx.x;
    if (i < e) atomicAdd(&deg[dst[i]], 1.0f);
}

__global__ void dinv_kernel(float* __restrict__ deg, int n) {
    int i = blockIdx.x * blockDim.x + threadIdx.x;
    if (i < n) deg[i] = rsqrtf(deg[i]);
}

// ---------------------------------------------------------------------------
// Aggregation: out[n] = bias + h[n]*dinv[n]^2 (self loop), then edge scatter.
// ---------------------------------------------------------------------------
__global__ __launch_bounds__(FEAT) void agg_init_kernel(
    const float* __restrict__ h, const float* __restrict__ dinv,
    const float* __restrict__ bias, float* __restrict__ out, int n) {
    int i = blockIdx.x;        // node
    int f = threadIdx.x;       // feature
    float di = dinv[i];
    out[(size_t)i * FEAT + f] = bias[f] + h[(size_t)i * FEAT + f] * di * di;
}

// 32 threads per edge, one float4 (4 features) per thread.
__global__ __launch_bounds__(256) void agg_scatter_kernel(
    const float* __restrict__ h, const int* __restrict__ src,
    const int* __restrict__ dst, const float* __restrict__ dinv,
    float* __restrict__ out, int E) {
    long long gid = (long long)blockIdx.x * blockDim.x + threadIdx.x;
    int e = (int)(gid >> 5);
    if (e >= E) return;
    int lane = (int)(gid & 31);
    int s = src[e], d = dst[e];
    float coef = dinv[s] * dinv[d];
    float4 v = ((const float4*)(h + (size_t)s * FEAT))[lane];
    float* op = out + (size_t)d * FEAT + lane * 4;
    atomicAdd(op + 0, v.x * coef);
    atomicAdd(op + 1, v.y * coef);
    atomicAdd(op + 2, v.z * coef);
    atomicAdd(op + 3, v.w * coef);
}

__global__ void relu_kernel(float* __restrict__ x, long long n) {
    long long i = (long long)blockIdx.x * blockDim.x + threadIdx.x;
    if (i < n) x[i] = fmaxf(x[i], 0.0f);
}

// ---------------------------------------------------------------------------
// Global mean pool into concat buffer zbuf[G][256]
// ---------------------------------------------------------------------------
__global__ void zero_kernel(float* __restrict__ p, int n) {
    int i = blockIdx.x * blockDim.x + threadIdx.x;
    if (i < n) p[i] = 0.0f;
}

__global__ __launch_bounds__(FEAT) void pool_acc_kernel(
    const float* __restrict__ h, const int* __restrict__ batch,
    float* __restrict__ zbuf, float* __restrict__ cnt, int side, int n) {
    int i = blockIdx.x;
    int f = threadIdx.x;
    int g = batch[i];
    atomicAdd(&zbuf[(size_t)g * 256 + side * FEAT + f], h[(size_t)i * FEAT + f]);
    if (f == 0) atomicAdd(&cnt[g], 1.0f);
}

__global__ __launch_bounds__(256) void pool_norm_kernel(
    float* __restrict__ zbuf, const float* __restrict__ counts, int G) {
    int g = blockIdx.x;
    int f = threadIdx.x;           // 0..255
    int side = f >> 7;
    float c = fmaxf(counts[side * G + g], 1.0f);
    zbuf[(size_t)g * 256 + f] /= c;
}

// ---------------------------------------------------------------------------
// Final head: out[G x 2] = z2[G x 256] @ Wc3[256 x 2] + bc3
// ---------------------------------------------------------------------------
__global__ void final_fc_kernel(const float* __restrict__ z,
                                const float* __restrict__ W,
                                const float* __restrict__ b,
                                float* __restrict__ out, int G) {
    int idx = blockIdx.x * blockDim.x + threadIdx.x;
    if (idx >= G * 2) return;
    int g = idx >> 1, o = idx & 1;
    const float* zp = z + (size_t)g * 256;
    float acc = b[o];
    for (int k = 0; k < 256; ++k) acc += zp[k] * W[k * 2 + o];
    out[idx] = acc;
}

// ---------------------------------------------------------------------------
extern "C" void kernel_launch(void* const* d_in, const int* in_sizes, int n_in,
                              void* d_out, int out_size, void* d_ws, size_t ws_size,
                              hipStream_t stream) {
    const float* x1     = (const float*)d_in[0];
    const float* x2     = (const float*)d_in[1];
    const int*   ei1    = (const int*)  d_in[2];
    const int*   ei2    = (const int*)  d_in[3];
    const int*   batch1 = (const int*)  d_in[4];
    const int*   batch2 = (const int*)  d_in[5];
    const float* W1  = (const float*)d_in[6];  const float* b1  = (const float*)d_in[7];
    const float* W2  = (const float*)d_in[8];  const float* b2  = (const float*)d_in[9];
    const float* W3  = (const float*)d_in[10]; const float* b3  = (const float*)d_in[11];
    const float* Wc1 = (const float*)d_in[12]; const float* bc1 = (const float*)d_in[13];
    const float* Wc2 = (const float*)d_in[14]; const float* bc2 = (const float*)d_in[15];
    const float* Wc3 = (const float*)d_in[16]; const float* bc3 = (const float*)d_in[17];

    const int N = in_sizes[0] / FEAT;   // 50000
    const int E = in_sizes[2] / 2;      // 500000
    const int G = out_size / 2;         // 512
    float* out = (float*)d_out;

    // workspace carve-up
    float* ws     = (float*)d_ws;
    float* dinv   = ws;  ws += N;
    float* buf0   = ws;  ws += (size_t)N * FEAT;   // GEMM output h
    float* buf1   = ws;  ws += (size_t)N * FEAT;   // aggregation output
    float* counts = ws;  ws += 2 * G;
    float* zbuf   = ws;  ws += (size_t)G * 256;    // concat [g1|g2]
    float* z1     = ws;  ws += (size_t)G * 512;
    float* z2     = ws;  ws += (size_t)G * 256;

    const int rt = (N + 15) / 16;       // row tiles for node GEMMs

    // zero pooled accumulators (poisoned workspace)
    zero_kernel<<<(G * 256 + 255) / 256, 256, 0, stream>>>(zbuf, G * 256);
    zero_kernel<<<(2 * G + 255) / 256, 256, 0, stream>>>(counts, 2 * G);

    for (int side = 0; side < 2; ++side) {
        const float* x     = side ? x2 : x1;
        const int*   src   = side ? ei2 : ei1;        // edge_index[0]
        const int*   dst   = src + E;                 // edge_index[1]
        const int*   batch = side ? batch2 : batch1;

        // degree (with self-loop) -> dinv
        deg_init_kernel <<<(N + 255) / 256, 256, 0, stream>>>(dinv, N);
        deg_count_kernel<<<(E + 255) / 256, 256, 0, stream>>>(dst, dinv, E);
        dinv_kernel     <<<(N + 255) / 256, 256, 0, stream>>>(dinv, N);

        const float* lw[3] = {W1, W2, W3};
        const float* lb[3] = {b1, b2, b3};
        const float* in = x;
        for (int l = 0; l < 3; ++l) {
            // h = in @ W_l
            gemm_wmma_f32<<<dim3(rt, 1), 256, 0, stream>>>(
                in, lw[l], nullptr, buf0, N, FEAT, FEAT, 0);
            // out = b_l + self-loop + edge scatter
            agg_init_kernel<<<N, FEAT, 0, stream>>>(buf0, dinv, lb[l], buf1, N);
            agg_scatter_kernel<<<(int)(((long long)E * 32 + 255) / 256), 256, 0, stream>>>(
                buf0, src, dst, dinv, buf1, E);
            if (l < 2)
                relu_kernel<<<(int)(((long long)N * FEAT + 255) / 256), 256, 0, stream>>>(
                    buf1, (long long)N * FEAT);
            in = buf1;
        }

        pool_acc_kernel<<<N, FEAT, 0, stream>>>(buf1, batch, zbuf, counts + side * G, side, N);
    }

    pool_norm_kernel<<<G, 256, 0, stream>>>(zbuf, counts, G);

    // MLP head: 256 -> 512 (relu) -> 256 (relu) -> 2
    gemm_wmma_f32<<<dim3(G / 16, 4), 256, 0, stream>>>(zbuf, Wc1, bc1, z1, G, 256, 512, 1);
    gemm_wmma_f32<<<dim3(G / 16, 2), 256, 0, stream>>>(z1,   Wc2, bc2, z2, G, 512, 256, 1);
    final_fc_kernel<<<(G * 2 + 255) / 256, 256, 0, stream>>>(z2, Wc3, bc3, out, G);
}